// KnnLoss_58377195487672
// MI455X (gfx1250) — compile-verified
//
#include <hip/hip_runtime.h>
#include <hip/hip_bf16.h>
#include <math.h>

typedef __attribute__((ext_vector_type(2))) float v2f;
typedef __attribute__((ext_vector_type(8))) float v8f;

#define K_NN    8
#define RADIUS  0.1f
#define C_CH    32

// One wave (32 lanes) per block. Block handles a 16-row tile of the distance
// matrix for one batch, sweeping all N/16 column tiles with
// V_WMMA_F32_16X16X4_F32:  d2 = A(-2x,-2y,-2z,1) * B(x,y,z,|q|^2) + C(|p|^2).
__global__ __launch_bounds__(32) void knn_loss_kernel(
    const float* __restrict__ pc,     // [B,N,3]
    const float* __restrict__ mask,   // [B,N,32]
    float* __restrict__ partial,      // [gridDim.x]
    int N, int tiles)
{
    __shared__ __align__(16) float s_tile[16 * 16];   // one 16x16 d2 tile
    __shared__ float s_sq[16];                        // row |p|^2
    __shared__ __align__(16) float s_topd[32 * K_NN]; // per-lane top-8 dists
    __shared__ int   s_topi[32 * K_NN];               // per-lane top-8 idx
    __shared__ int   s_fidx[16 * K_NN];               // final idx per row

    const int lane = threadIdx.x;      // 0..31
    const int m    = lane & 15;        // row/col within tile (A/B operand slot)
    const int hi   = lane >> 4;        // selects K-pair (A/B) & M-half (C/D)
    const int b    = blockIdx.x / tiles;
    const int rt   = blockIdx.x % tiles;
    const int row_base = rt * 16;

    const float* pcB   = pc   + (size_t)b * N * 3;
    const float* maskB = mask + (size_t)b * N * C_CH;

    // ---- A operand: lane holds A[m][k0], A[m][k0+1] with k0 = 2*hi ----
    const float* rp = pcB + (size_t)(row_base + m) * 3;
    float rx = rp[0], ry = rp[1], rz = rp[2];
    if (hi == 0) s_sq[m] = rx * rx + ry * ry + rz * rz;
    __syncthreads();

    v2f a;
    a.x = hi ? (-2.0f * rz) : (-2.0f * rx);
    a.y = hi ? 1.0f         : (-2.0f * ry);

    // ---- C operand: c[r] = |p_{row_base + r + 8*hi}|^2 (constant per lane) ----
    v8f cinit;
#pragma unroll
    for (int r = 0; r < 8; ++r) cinit[r] = s_sq[r + 8 * hi];

    // ---- per-lane top-8 state: lane pair (2*row, 2*row+1) covers one row ----
    float dv[K_NN];
    int   iv[K_NN];
#pragma unroll
    for (int k = 0; k < K_NN; ++k) { dv[k] = 3.4e38f; iv[k] = 0; }

    // Branchless accept path: only the reject test is a branch (whole-wave
    // skip via v_cmpx + s_cbranch_execz); the bubble pass is pure cndmask
    // chains so it dual-issues on the wave32 VALU.
    auto ins = [&](float d, int gi) {
        if (d < dv[K_NN - 1]) {
            dv[K_NN - 1] = d; iv[K_NN - 1] = gi;
#pragma unroll
            for (int s = K_NN - 1; s > 0; --s) {
                const bool sw = dv[s] < dv[s - 1]; // ties keep earlier idx
                const float lo = sw ? dv[s]     : dv[s - 1];
                const float h2 = sw ? dv[s - 1] : dv[s];
                const int   li = sw ? iv[s]     : iv[s - 1];
                const int   h3 = sw ? iv[s - 1] : iv[s];
                dv[s - 1] = lo; dv[s] = h2;
                iv[s - 1] = li; iv[s] = h3;
            }
        }
    };

    const int rrow = lane >> 1;    // 0..15: row this lane helps select for
    const int half = lane & 1;     // which 8 columns of the tile

    // Software-pipelined column-point load: fetch tile j+1 while computing j.
    float qx = pcB[(size_t)m * 3 + 0];
    float qy = pcB[(size_t)m * 3 + 1];
    float qz = pcB[(size_t)m * 3 + 2];

    for (int j = 0; j < tiles; ++j) {
        const float cx = qx, cy = qy, cz = qz;

        if (j + 1 < tiles) {                       // issue next tile's load now
            const float* np = pcB + (size_t)((j + 1) * 16 + m) * 3;
            qx = np[0]; qy = np[1]; qz = np[2];
        }
        if (j + 8 < tiles)                         // warm L2/WGP$ further out
            __builtin_prefetch(pcB + (size_t)((j + 8) * 16 + m) * 3, 0, 1);

        // B operand: lane holds B[k0][n], B[k0+1][n], k0 = 2*hi
        const float qsq = cx * cx + cy * cy + cz * cz;
        v2f bv;
        bv.x = hi ? cz  : cx;
        bv.y = hi ? qsq : cy;

        // d[r] = d2 for row (r + 8*hi), column (lane&15) of this tile
        v8f d = __builtin_amdgcn_wmma_f32_16x16x4_f32(
            false, a, false, bv, (short)0, cinit, false, false);

        __syncthreads();   // previous iteration's tile reads complete
#pragma unroll
        for (int r = 0; r < 8; ++r) {
            const float t = d[r];
            s_tile[(r + 8 * hi) * 16 + m] = (t < 0.0f) ? 0.0f : t;
        }
        __syncthreads();   // tile visible to repartitioned lanes

        // top-8 update: lane scans 8 columns of its row (increasing index)
        const float4* q4 = (const float4*)&s_tile[rrow * 16 + half * 8];
        const float4 c0 = q4[0];
        const float4 c1 = q4[1];
        const int g = j * 16 + half * 8;
        ins(c0.x, g + 0); ins(c0.y, g + 1); ins(c0.z, g + 2); ins(c0.w, g + 3);
        ins(c1.x, g + 4); ins(c1.y, g + 5); ins(c1.z, g + 6); ins(c1.w, g + 7);
    }

    // ---- merge the two sorted lists per row, apply radius rule ----
#pragma unroll
    for (int k = 0; k < K_NN; ++k) {
        s_topd[lane * K_NN + k] = dv[k];
        s_topi[lane * K_NN + k] = iv[k];
    }
    __syncthreads();

    if (lane < 16) {
        const float* da = &s_topd[(2 * lane)     * K_NN];
        const float* db = &s_topd[(2 * lane + 1) * K_NN];
        const int*   ja = &s_topi[(2 * lane)     * K_NN];
        const int*   jb = &s_topi[(2 * lane + 1) * K_NN];
        int pa = 0, pb = 0, selfi = 0;
#pragma unroll
        for (int k = 0; k < K_NN; ++k) {
            const float xa = da[pa], xb = db[pb];
            const int   ia = ja[pa], ib = jb[pb];
            const bool tA = (xa < xb) || (xa == xb && ia < ib); // stable
            const float dsel = tA ? xa : xb;
            const int   isel = tA ? ia : ib;
            pa += tA ? 1 : 0;
            pb += tA ? 0 : 1;
            if (k == 0) selfi = isel;               // self_idx = nearest
            const float eu = sqrtf(dsel);
            int fi = (eu > RADIUS) ? selfi : isel;  // radius replacement
            fi = fi < 0 ? 0 : (fi > N - 1 ? N - 1 : fi);
            s_fidx[lane * K_NN + k] = fi;
        }
    }
    __syncthreads();

    // ---- loss: lane = channel (C=32 == wave32), coalesced 128B mask rows ----
    float acc = 0.0f;
    for (int r = 0; r < 16; ++r) {
        const int n = row_base + r;
        const float mref = maskB[(size_t)n * C_CH + lane];
#pragma unroll
        for (int k = 0; k < K_NN; ++k) {
            const int fj = s_fidx[r * K_NN + k];
            const float mj = maskB[(size_t)fj * C_CH + lane];
            acc += fabsf(mref - mj);
        }
    }
#pragma unroll
    for (int off = 16; off > 0; off >>= 1)
        acc += __shfl_xor(acc, off, 32);
    if (lane == 0) partial[blockIdx.x] = acc;
}

// Deterministic fixed-order tree reduction of per-block partials.
__global__ __launch_bounds__(256) void knn_reduce_kernel(
    const float* __restrict__ partial, float* __restrict__ out,
    int count, float scale)
{
    __shared__ float s[256];
    const int t = threadIdx.x;
    float a = 0.0f;
    for (int i = t; i < count; i += 256) a += partial[i];
    s[t] = a;
    __syncthreads();
#pragma unroll
    for (int w = 128; w > 0; w >>= 1) {
        if (t < w) s[t] += s[t + w];
        __syncthreads();
    }
    if (t == 0) out[0] = s[0] * scale;
}

extern "C" void kernel_launch(void* const* d_in, const int* in_sizes, int n_in,
                              void* d_out, int out_size, void* d_ws, size_t ws_size,
                              hipStream_t stream) {
    const float* pc   = (const float*)d_in[0];   // [B,N,3] float32
    const float* mask = (const float*)d_in[1];   // [B,N,32] float32
    float* out = (float*)d_out;                  // scalar float32

    const int B = 4;
    const int N = in_sizes[0] / (3 * B);         // 8192
    const int tiles = N / 16;                    // 512
    const int grid = B * tiles;                  // 2048 blocks

    float* partial = (float*)d_ws;               // grid * 4 bytes of scratch

    knn_loss_kernel<<<grid, 32, 0, stream>>>(pc, mask, partial, N, tiles);

    const float scale = 1.0f / ((float)B * (float)N * (float)K_NN);
    knn_reduce_kernel<<<1, 256, 0, stream>>>(partial, out, grid, scale);
}